// FocalLoss_62749472195356
// MI455X (gfx1250) — compile-verified
//
#include <hip/hip_runtime.h>
#include <stdint.h>

// Problem constants (from reference)
#define NB 4
#define NA 120000
#define NC 80
#define NM 32
#define APB 256                       // anchors per block
#define TPB 256                       // threads per block (8 wave32)
#define NBLK ((NA + APB - 1) / APB)   // 469 blocks per batch

typedef float v4f __attribute__((ext_vector_type(4)));

// Exact parameter type reported by clang for the async-LDS builtins:
// 'int __attribute__((vector_size(16)))' with __device__ (AS1) / __shared__ (AS3)
typedef int v4i_b __attribute__((vector_size(4 * sizeof(int))));
typedef __attribute__((address_space(1))) v4i_b* as1_v4i;
typedef __attribute__((address_space(3))) v4i_b* as3_v4i;

// ---- CDNA5 async global->LDS copy (16B per lane) -------------------------
__device__ __forceinline__ void async_cp16(const float* g, float* s) {
#if __has_builtin(__builtin_amdgcn_global_load_async_to_lds_b128)
  __builtin_amdgcn_global_load_async_to_lds_b128((as1_v4i)g, (as3_v4i)s, 0, 0);
#else
  // VDST = LDS byte address (low 32 bits of flat LDS pointer), VADDR = 64-bit global addr
  unsigned l = (unsigned)(uintptr_t)s;
  unsigned long long ga = (unsigned long long)(uintptr_t)g;
  asm volatile("global_load_async_to_lds_b128 %0, %1, off"
               :: "v"(l), "v"(ga) : "memory");
#endif
}

__device__ __forceinline__ void wait_async0() {
#if __has_builtin(__builtin_amdgcn_s_wait_asynccnt)
  __builtin_amdgcn_s_wait_asynccnt(0);
#else
  asm volatile("s_wait_asynccnt 0" ::: "memory");
#endif
}

// ---- main fused kernel ---------------------------------------------------
__global__ __launch_bounds__(TPB) void focal_main(
    const float* __restrict__ cls,   // (B, A, C)
    const float* __restrict__ reg,   // (B, A, 8)
    const float* __restrict__ anc,   // (1, A, 4)
    const float* __restrict__ ann,   // (B, M, 10)
    float* __restrict__ ws)          // [3][NB][NBLK] partial sums
{
  __shared__ __align__(16) float s_ann[NM * 10];  // 320 floats
  __shared__ int   s_meta[APB];                   // state<<8 | assigned_cls
  __shared__ float s_red[3][TPB / 32];

  const int tid = threadIdx.x;
  const int b   = blockIdx.y;
  const int a0  = blockIdx.x * APB;

  // Phase 0: DMA this batch's annotation table into LDS (80 x 16B).
  if (tid < (NM * 10) / 4)
    async_cp16(ann + (size_t)b * NM * 10 + (size_t)tid * 4, &s_ann[tid * 4]);
  wait_async0();
  __syncthreads();

  float cls_p = 0.f, reg_p = 0.f, pos_p = 0.f;

  // Phase 1: one thread per anchor -> IoU argmax, state, reg loss.
  const int a = a0 + tid;
  int state = 3, acls = 0;   // 0=neg 1=pos 2=ignore 3=out-of-range
  if (a < NA) {
    v4f ab = *(const v4f*)(anc + (size_t)a * 4);
    float aw  = ab.z - ab.x, ah = ab.w - ab.y;
    float acx = ab.x + 0.5f * aw, acy = ab.y + 0.5f * ah;
    float area_a = aw * ah;

    float best = -1.f; int arg = 0;
    for (int m = 0; m < NM; ++m) {
      float bx1 = s_ann[m * 10 + 4], by1 = s_ann[m * 10 + 5];
      float bx2 = s_ann[m * 10 + 6], by2 = s_ann[m * 10 + 7];
      float iw = fmaxf(fminf(ab.z, bx2) - fmaxf(ab.x, bx1), 0.f);
      float ih = fmaxf(fminf(ab.w, by2) - fmaxf(ab.y, by1), 0.f);
      float inter = iw * ih;
      float ua = fmaxf(area_a + (bx2 - bx1) * (by2 - by1) - inter, 1e-8f);
      float iou = inter / ua;
      if (iou > best) { best = iou; arg = m; }   // first-max, like jnp.argmax
    }
    bool pos = best >= 0.5f;
    state = pos ? 1 : (best < 0.4f ? 0 : 2);
    acls  = (int)s_ann[arg * 10 + 8];

    if (pos) {
      const float* r = &s_ann[arg * 10];
      float gw_f = r[6] - r[4], gh_f = r[7] - r[5];
      float gcx_f = r[4] + 0.5f * gw_f, gcy_f = r[5] + 0.5f * gh_f; // centers before clip
      gw_f = fmaxf(gw_f, 1.f); gh_f = fmaxf(gh_f, 1.f);
      float t0 = (gcx_f - acx) / aw * 10.f;   // / STD 0.1
      float t1 = (gcy_f - acy) / ah * 10.f;
      float t2 = __logf(gw_f / aw) * 5.f;     // / STD 0.2
      float t3 = __logf(gh_f / ah) * 5.f;
      v4f rg = *(const v4f*)(reg + ((size_t)b * NA + a) * 8);
      float d[4] = { fabsf(t0 - rg.x), fabsf(t1 - rg.y),
                     fabsf(t2 - rg.z), fabsf(t3 - rg.w) };
      float rl = 0.f;
      #pragma unroll
      for (int k = 0; k < 4; ++k)
        rl += (d[k] <= (1.f / 9.f)) ? 4.5f * d[k] * d[k] : d[k] - (0.5f / 9.f);
      reg_p = rl;
      pos_p = 1.f;
    }
  }
  s_meta[tid] = acls | (state << 8);
  __syncthreads();

  // Phase 2: coalesced NT streaming of this block's classification slab.
  const int nA = (NA - a0 < APB) ? (NA - a0) : APB;
  const v4f* cls4 = (const v4f*)(cls + ((size_t)b * NA + (size_t)a0) * NC);
  const int n4 = nA * (NC / 4);   // 20 float4 per anchor
  for (int i = tid; i < n4; i += TPB) {
    int al = i / (NC / 4);
    int c0 = (i % (NC / 4)) * 4;
    int meta = s_meta[al];
    int st = meta >> 8;
    if (st == 2) continue;        // ignored anchor: targets == -1 -> 0 loss
    int ac = meta & 0xff;
#if __has_builtin(__builtin_nontemporal_load)
    v4f p4 = __builtin_nontemporal_load(cls4 + i);
#else
    v4f p4 = cls4[i];
#endif
    #pragma unroll
    for (int j = 0; j < 4; ++j) {
      float p = fminf(fmaxf(p4[j], 1e-4f), 1.f - 1e-4f);
      if (st == 1 && (c0 + j) == ac) {
        float q = 1.f - p;
        cls_p += 0.75f * q * q * (-__logf(p));       // target == 1
      } else {
        cls_p += 0.25f * p * p * (-__logf(1.f - p)); // target == 0
      }
    }
  }

  // Phase 3: wave32 shfl reduction -> cross-wave LDS -> deterministic slot write.
  #pragma unroll
  for (int off = 16; off > 0; off >>= 1) {
    cls_p += __shfl_xor(cls_p, off, 32);
    reg_p += __shfl_xor(reg_p, off, 32);
    pos_p += __shfl_xor(pos_p, off, 32);
  }
  const int wave = tid >> 5, lane = tid & 31;
  if (lane == 0) { s_red[0][wave] = cls_p; s_red[1][wave] = reg_p; s_red[2][wave] = pos_p; }
  __syncthreads();
  if (tid == 0) {
    float c = 0.f, r = 0.f, n = 0.f;
    #pragma unroll
    for (int w = 0; w < TPB / 32; ++w) { c += s_red[0][w]; r += s_red[1][w]; n += s_red[2][w]; }
    ws[(0 * NB + b) * NBLK + blockIdx.x] = c;
    ws[(1 * NB + b) * NBLK + blockIdx.x] = r;
    ws[(2 * NB + b) * NBLK + blockIdx.x] = n;
  }
}

// ---- deterministic finalize: 12 waves, one (quantity,batch) pair each ----
__global__ __launch_bounds__(384) void focal_final(const float* __restrict__ ws,
                                                   float* __restrict__ out) {
  __shared__ float s_sum[12];
  const int w = threadIdx.x >> 5, lane = threadIdx.x & 31;
  const float* src = ws + (size_t)w * NBLK;
  float s = 0.f;
  for (int i = lane; i < NBLK; i += 32) s += src[i];
  #pragma unroll
  for (int off = 16; off > 0; off >>= 1) s += __shfl_xor(s, off, 32);
  if (lane == 0) s_sum[w] = s;
  __syncthreads();
  if (threadIdx.x == 0) {
    float cm = 0.f, rm = 0.f;
    for (int b = 0; b < NB; ++b) {
      float csum = s_sum[0 * NB + b];
      float rsum = s_sum[1 * NB + b];
      float np   = s_sum[2 * NB + b];
      float npc  = fmaxf(np, 1.f);
      cm += csum / npc;
      rm += (np > 0.f) ? rsum / (npc * 4.f) : 0.f;
    }
    out[0] = cm * (1.f / NB);
    out[1] = rm * (1.f / NB);
  }
}

extern "C" void kernel_launch(void* const* d_in, const int* in_sizes, int n_in,
                              void* d_out, int out_size, void* d_ws, size_t ws_size,
                              hipStream_t stream) {
  const float* cls = (const float*)d_in[0];  // (B, A, C) f32
  const float* reg = (const float*)d_in[1];  // (B, A, 8) f32
  const float* anc = (const float*)d_in[2];  // (1, A, 4) f32
  const float* ann = (const float*)d_in[3];  // (B, M, 10) f32
  float* ws  = (float*)d_ws;                 // needs 3*NB*NBLK*4 = ~22.5 KB
  float* out = (float*)d_out;                // 2 floats

  dim3 grid(NBLK, NB);
  focal_main<<<grid, TPB, 0, stream>>>(cls, reg, anc, ann, ws);
  focal_final<<<1, 384, 0, stream>>>(ws, out);
}